// DecoderMulti_7078106104338
// MI455X (gfx1250) — compile-verified
//
#include <hip/hip_runtime.h>
#include <hip/hip_bf16.h>
#include <math.h>

// ---------------------------------------------------------------------------
// Problem constants (from reference)
// ---------------------------------------------------------------------------
#define TT   64          // max_trg_len
#define BB   256         // batch
#define HID  512
#define LL   64          // route length
#define GDIM 1536        // 3*HID
#define KIN  1090        // RNN_IN
#define FC1  1024        // 2*HID
#define STEPS 63         // T-1 scan steps
#define FEA  32          // rid feature width

typedef float v2f __attribute__((ext_vector_type(2)));
typedef float v8f __attribute__((ext_vector_type(8)));

__device__ __forceinline__ v8f wmma4(v2f a, v2f b, v8f c) {
  // D = A(16x4 f32) * B(4x16 f32) + C(16x16 f32)
  return __builtin_amdgcn_wmma_f32_16x16x4_f32(
      false, a, false, b, (short)0, c, false, false);
}

__device__ __forceinline__ float sigmoidf_(float x) {
  return 1.0f / (1.0f + __expf(-x));
}
__device__ __forceinline__ v2f ldv2(const float* p) {
  return *(const v2f*)p;   // all call sites are 8B-aligned
}

// ---------------------------------------------------------------------------
// Zero the output buffer (row 0 of both outputs must be zero).
// ---------------------------------------------------------------------------
__global__ void zero_kernel(float* p, int n) {
  int i = blockIdx.x * blockDim.x + threadIdx.x;
  if (i < n) p[i] = 0.0f;
}

// ---------------------------------------------------------------------------
// Prep: W25[n,c] = W_ih[n,512+c] + W_ih[n,1057+c]  (c<32)
//       W4c[n,k] = W_ih[n,545+k]                   (k<512)  -> 8B-aligned GEMM B
// ---------------------------------------------------------------------------
__global__ void prep_kernel(const float* __restrict__ W_ih,
                            float* __restrict__ w25,
                            float* __restrict__ w4c) {
  int i = blockIdx.x * blockDim.x + threadIdx.x;
  if (i >= GDIM * (FEA + HID)) return;
  int n = i / (FEA + HID);
  int c = i - n * (FEA + HID);
  const float* row = W_ih + (size_t)n * KIN;
  if (c < FEA) {
    w25[(size_t)n * FEA + c] = row[512 + c] + row[1057 + c];
  } else {
    int k = c - FEA;
    w4c[(size_t)n * HID + k] = row[545 + k];
  }
}

// ---------------------------------------------------------------------------
// dcontrib[b,g] = emb_id[d_rids[b]] @ W4^T + d_rates[b]*w6[g] + b_ih[g]
// WMMA GEMM M=256, N=1536, K=512; 32 rows per wave (2 acc chains share B).
// grid = (96 n-tiles, 8 m-tiles), block = 32.
// ---------------------------------------------------------------------------
__global__ void dcontrib_kernel(const int* __restrict__ d_rids,
                                const float* __restrict__ d_rates,
                                const float* __restrict__ emb,
                                const float* __restrict__ w4c,
                                const float* __restrict__ W_ih,
                                const float* __restrict__ b_ih,
                                float* __restrict__ dcon) {
  const int lane = threadIdx.x;
  const int lo = lane & 15, hi = lane >> 4;
  const int n0 = blockIdx.x * 16;
  const int m0 = blockIdx.y * 32;

  const float* e0 = emb + (size_t)d_rids[m0 + lo] * HID;
  const float* e1 = emb + (size_t)d_rids[m0 + 16 + lo] * HID;
  const float* wrow = w4c + (size_t)(n0 + lo) * HID;

  v8f acc0 = {}, acc1 = {};
#pragma unroll 4
  for (int k0 = 0; k0 < HID; k0 += 4) {
    const int ka = k0 + 2 * hi;
    const v2f bv = ldv2(wrow + ka);
    acc0 = wmma4(ldv2(e0 + ka), bv, acc0);
    acc1 = wmma4(ldv2(e1 + ka), bv, acc1);
  }

  const int col = n0 + lo;
  const float w6c = W_ih[(size_t)col * KIN + 1089];
  const float bi = b_ih[col];
#pragma unroll
  for (int v = 0; v < 8; ++v) {
    const int r0 = m0 + v + 8 * hi;
    const int r1 = r0 + 16;
    dcon[(size_t)r0 * GDIM + col] = acc0[v] + d_rates[r0] * w6c + bi;
    dcon[(size_t)r1 * GDIM + col] = acc1[v] + d_rates[r1] * w6c + bi;
  }
}

// ---------------------------------------------------------------------------
// gates_x[m,g] = emb_id[id[m]]@W1^T + feats[id[m]]@W25^T + xr[m]*w3[g]
//               + dcontrib[b(m),g]          (bias already folded in dcontrib)
// M=16128, N=1536; K=512 then K=32; 32 rows per wave.
// grid = (96 n-tiles, 504 m-tiles), block = 32.
// ---------------------------------------------------------------------------
__global__ void gates_kernel(const int* __restrict__ trg_id,
                             const float* __restrict__ trg_rate,
                             const float* __restrict__ emb,
                             const float* __restrict__ fd,
                             const float* __restrict__ W_ih,
                             const float* __restrict__ w25,
                             const float* __restrict__ dcon,
                             float* __restrict__ gates) {
  const int lane = threadIdx.x;
  const int lo = lane & 15, hi = lane >> 4;
  const int n0 = blockIdx.x * 16;
  const int m0 = blockIdx.y * 32;

  const int id0 = trg_id[m0 + lo];
  const int id1 = trg_id[m0 + 16 + lo];
  const float* e0 = emb + (size_t)id0 * HID;
  const float* e1 = emb + (size_t)id1 * HID;
  const float* f0 = fd + (size_t)id0 * FEA;
  const float* f1 = fd + (size_t)id1 * FEA;
  const float* wrow = W_ih + (size_t)(n0 + lo) * KIN;   // W1 cols 0..511
  const float* w25row = w25 + (size_t)(n0 + lo) * FEA;

  v8f acc0 = {}, acc1 = {};
#pragma unroll 4
  for (int k0 = 0; k0 < HID; k0 += 4) {     // emb segment, K=512
    const int ka = k0 + 2 * hi;
    const v2f bv = ldv2(wrow + ka);
    acc0 = wmma4(ldv2(e0 + ka), bv, acc0);
    acc1 = wmma4(ldv2(e1 + ka), bv, acc1);
  }
#pragma unroll
  for (int k0 = 0; k0 < FEA; k0 += 4) {     // feats segment, K=32 (W2+W5)
    const int ka = k0 + 2 * hi;
    const v2f bv = ldv2(w25row + ka);
    acc0 = wmma4(ldv2(f0 + ka), bv, acc0);
    acc1 = wmma4(ldv2(f1 + ka), bv, acc1);
  }

  const int col = n0 + lo;
  const float w3c = W_ih[(size_t)col * KIN + 544];
#pragma unroll
  for (int v = 0; v < 8; ++v) {
    const int r0 = m0 + v + 8 * hi;
    const int r1 = r0 + 16;
    gates[(size_t)r0 * GDIM + col] =
        acc0[v] + trg_rate[r0] * w3c + dcon[(size_t)(r0 & 255) * GDIM + col];
    gates[(size_t)r1 * GDIM + col] =
        acc1[v] + trg_rate[r1] * w3c + dcon[(size_t)(r1 & 255) * GDIM + col];
  }
}

// ---------------------------------------------------------------------------
// Per-step GRU: gh = h @ W_hh^T + b_hh ; fused r,z,n gates -> h_new.
// One wave per 16(b) x 16(j) tile; 3 WMMA chains share the A loads.
// grid = (32 j-tiles, 16 b-tiles), block = 32.
// ---------------------------------------------------------------------------
__global__ void gru_kernel(const float* __restrict__ gx,    // B x 1536
                           const float* __restrict__ h_cur, // B x 512
                           const float* __restrict__ W_hh,  // 1536 x 512
                           const float* __restrict__ b_hh,
                           float* __restrict__ h_new) {
  const int lane = threadIdx.x;
  const int lo = lane & 15, hi = lane >> 4;
  const int j0 = blockIdx.x * 16;
  const int b0 = blockIdx.y * 16;

  const float* hrow = h_cur + (size_t)(b0 + lo) * HID;
  const float* wr = W_hh + (size_t)(j0 + lo) * HID;
  const float* wz = W_hh + (size_t)(512 + j0 + lo) * HID;
  const float* wn = W_hh + (size_t)(1024 + j0 + lo) * HID;

  v8f ar = {}, az = {}, an = {};
#pragma unroll 2
  for (int k0 = 0; k0 < HID; k0 += 4) {
    const int ka = k0 + 2 * hi;
    const v2f a = ldv2(hrow + ka);
    ar = wmma4(a, ldv2(wr + ka), ar);
    az = wmma4(a, ldv2(wz + ka), az);
    an = wmma4(a, ldv2(wn + ka), an);
  }

  const int j = j0 + lo;
  const float br = b_hh[j], bz = b_hh[512 + j], bn = b_hh[1024 + j];
#pragma unroll
  for (int v = 0; v < 8; ++v) {
    const int bi = b0 + v + 8 * hi;
    const size_t base = (size_t)bi * GDIM;
    const float r  = sigmoidf_(gx[base + j]       + ar[v] + br);
    const float z  = sigmoidf_(gx[base + 512 + j] + az[v] + bz);
    const float nn = tanhf(gx[base + 1024 + j] + r * (an[v] + bn));
    const float hp = h_cur[(size_t)bi * HID + j];
    h_new[(size_t)bi * HID + j] = (1.0f - z) * nn + z * hp;
  }
}

// ---------------------------------------------------------------------------
// Per-step attention: energies -> masked softmax -> pred_id + weighted sum.
// One block (256 threads) per batch element.  ~1% of FLOPs -> plain VALU.
// ---------------------------------------------------------------------------
__global__ void attn_kernel(const float* __restrict__ h_new,
                            const float* __restrict__ route_outputs, // (L,B,H)
                            const int* __restrict__ mask,            // (B,L)
                            float* __restrict__ out_id,              // (T,B,L)
                            float* __restrict__ weighted,            // (B,H)
                            int t) {
  const int b = blockIdx.x;
  const int tid = threadIdx.x;
  __shared__ float sh_h[HID];
  __shared__ float sh_part[256];
  __shared__ float sh_e[LL];
  __shared__ float sh_s[LL];
  __shared__ float sh_red[2];

  sh_h[tid]       = h_new[(size_t)b * HID + tid];
  sh_h[tid + 256] = h_new[(size_t)b * HID + tid + 256];
  __syncthreads();

  {
    const int l = tid >> 2, part = tid & 3;
    const float* key = route_outputs + ((size_t)l * BB + b) * HID;
    float p = 0.0f;
    const int kb = part * 128;
    for (int k = kb; k < kb + 128; ++k) p += sh_h[k] * key[k];
    sh_part[tid] = p;
  }
  __syncthreads();

  if (tid < LL) {
    float e = (sh_part[4 * tid] + sh_part[4 * tid + 1] +
               sh_part[4 * tid + 2] + sh_part[4 * tid + 3]) * 0.04419417382f;
    sh_e[tid] = (mask[b * LL + tid] == 0) ? -1.0e10f : e;
  }
  __syncthreads();

  if (tid == 0) {
    float mx = sh_e[0];
    for (int i = 1; i < LL; ++i) mx = fmaxf(mx, sh_e[i]);
    float sm = 0.0f;
    for (int i = 0; i < LL; ++i) sm += __expf(sh_e[i] - mx);
    sh_red[0] = mx;
    sh_red[1] = sm;
  }
  __syncthreads();

  if (tid < LL) {
    const float s = __expf(sh_e[tid] - sh_red[0]) / sh_red[1];
    sh_s[tid] = s;
    out_id[(((size_t)(t + 1)) * BB + b) * LL + tid] =
        (mask[b * LL + tid] == 0) ? 0.0f : s;
  }
  __syncthreads();

#pragma unroll
  for (int rep = 0; rep < 2; ++rep) {
    const int h = tid + rep * 256;
    float acc = 0.0f;
    for (int l = 0; l < LL; ++l)
      acc += sh_s[l] * route_outputs[((size_t)l * BB + b) * HID + h];
    weighted[(size_t)b * HID + h] = acc;
  }
}

// ---------------------------------------------------------------------------
// Per-step fc1: hid1 = relu([h_new, weighted] @ W_fc1^T + b_fc1)
// WMMA GEMM M=256, N=1024, K=1024; 32 rows per wave.
// grid = (64 n-tiles, 8 b-tiles), block = 32.
// ---------------------------------------------------------------------------
__global__ void fc1_kernel(const float* __restrict__ h_new,
                           const float* __restrict__ wtd,
                           const float* __restrict__ W_fc1, // 1024 x 1024
                           const float* __restrict__ b_fc1,
                           float* __restrict__ hid1) {
  const int lane = threadIdx.x;
  const int lo = lane & 15, hi = lane >> 4;
  const int n0 = blockIdx.x * 16;
  const int b0 = blockIdx.y * 32;

  const float* ha0 = h_new + (size_t)(b0 + lo) * HID;
  const float* ha1 = h_new + (size_t)(b0 + 16 + lo) * HID;
  const float* wa0 = wtd + (size_t)(b0 + lo) * HID;
  const float* wa1 = wtd + (size_t)(b0 + 16 + lo) * HID;
  const float* wrow = W_fc1 + (size_t)(n0 + lo) * FC1;

  v8f acc0 = {}, acc1 = {};
#pragma unroll 4
  for (int k0 = 0; k0 < FC1; k0 += 4) {
    const int ka = k0 + 2 * hi;
    // segment boundary (512) is 4-aligned: the pair never straddles it
    const float* s0 = (ka < HID) ? (ha0 + ka) : (wa0 + (ka - HID));
    const float* s1 = (ka < HID) ? (ha1 + ka) : (wa1 + (ka - HID));
    const v2f bv = ldv2(wrow + ka);
    acc0 = wmma4(ldv2(s0), bv, acc0);
    acc1 = wmma4(ldv2(s1), bv, acc1);
  }

  const int col = n0 + lo;
  const float bias = b_fc1[col];
#pragma unroll
  for (int v = 0; v < 8; ++v) {
    const int r0 = b0 + v + 8 * hi;
    const int r1 = r0 + 16;
    hid1[(size_t)r0 * FC1 + col] = fmaxf(acc0[v] + bias, 0.0f);
    hid1[(size_t)r1 * FC1 + col] = fmaxf(acc1[v] + bias, 0.0f);
  }
}

// ---------------------------------------------------------------------------
// Per-step fc2: pred_rate[b] = sigmoid(hid1[b,:].W_fc2 + b_fc2), length-masked.
// 8 waves/block, one wave per batch element.
// ---------------------------------------------------------------------------
__global__ void fc2_kernel(const float* __restrict__ hid1,
                           const float* __restrict__ W_fc2, // 1 x 1024
                           const float* __restrict__ b_fc2,
                           const int* __restrict__ trg_len,
                           float* __restrict__ out_rate,    // (T,B)
                           int t) {
  const int wave = threadIdx.x >> 5;
  const int lane = threadIdx.x & 31;
  const int b = blockIdx.x * 8 + wave;
  const float* row = hid1 + (size_t)b * FC1;
  float acc = 0.0f;
  for (int k = lane; k < FC1; k += 32) acc += row[k] * W_fc2[k];
#pragma unroll
  for (int o = 16; o > 0; o >>= 1) acc += __shfl_xor(acc, o, 32);
  if (lane == 0) {
    const float v = sigmoidf_(acc + b_fc2[0]);
    const bool valid = (t + 1) < trg_len[b];
    out_rate[(size_t)(t + 1) * BB + b] = valid ? v : 0.0f;
  }
}

// ---------------------------------------------------------------------------
// Host launcher
// ---------------------------------------------------------------------------
extern "C" void kernel_launch(void* const* d_in, const int* in_sizes, int n_in,
                              void* d_out, int out_size, void* d_ws,
                              size_t ws_size, hipStream_t stream) {
  const int*   trg_id        = (const int*)  d_in[2];
  const float* trg_rate      = (const float*)d_in[3];
  const int*   trg_len       = (const int*)  d_in[4];
  const float* hidden        = (const float*)d_in[5];
  const float* rid_features  = (const float*)d_in[6];
  const float* route_outputs = (const float*)d_in[8];
  const int*   route_mask    = (const int*)  d_in[9];
  const int*   d_rids        = (const int*)  d_in[10];
  const float* d_rates       = (const float*)d_in[11];
  const float* emb_id        = (const float*)d_in[13];
  const float* W_ih          = (const float*)d_in[14];
  const float* W_hh          = (const float*)d_in[15];
  const float* b_ih          = (const float*)d_in[16];
  const float* b_hh          = (const float*)d_in[17];
  const float* W_fc1         = (const float*)d_in[18];
  const float* b_fc1         = (const float*)d_in[19];
  const float* W_fc2         = (const float*)d_in[20];
  const float* b_fc2         = (const float*)d_in[21];

  float* out_id   = (float*)d_out;                 // (T,B,L)
  float* out_rate = out_id + (size_t)TT * BB * LL; // (T,B,1)

  // Workspace layout (floats)
  float* ws    = (float*)d_ws;
  float* gates = ws;                                   // 63*256*1536
  float* hbuf0 = gates + (size_t)STEPS * BB * GDIM;    // 256*512
  float* hbuf1 = hbuf0 + (size_t)BB * HID;             // 256*512
  float* wtd   = hbuf1 + (size_t)BB * HID;             // 256*512
  float* hid1  = wtd   + (size_t)BB * HID;             // 256*1024
  float* w25   = hid1  + (size_t)BB * FC1;             // 1536*32
  float* w4c   = w25   + (size_t)GDIM * FEA;           // 1536*512
  float* dcon  = w4c   + (size_t)GDIM * HID;           // 256*1536

  {
    const int n = TT * BB * LL + TT * BB;
    zero_kernel<<<(n + 255) / 256, 256, 0, stream>>>(out_id, n);
  }

  hipMemcpyAsync(hbuf0, hidden, (size_t)BB * HID * sizeof(float),
                 hipMemcpyDeviceToDevice, stream);

  {
    const int n = GDIM * (FEA + HID);
    prep_kernel<<<(n + 255) / 256, 256, 0, stream>>>(W_ih, w25, w4c);
  }

  dcontrib_kernel<<<dim3(GDIM / 16, BB / 32), 32, 0, stream>>>(
      d_rids, d_rates, emb_id, w4c, W_ih, b_ih, dcon);

  gates_kernel<<<dim3(GDIM / 16, STEPS * BB / 32), 32, 0, stream>>>(
      trg_id, trg_rate, emb_id, rid_features, W_ih, w25, dcon, gates);

  for (int t = 0; t < STEPS; ++t) {
    const float* hc = (t & 1) ? hbuf1 : hbuf0;
    float*       hn = (t & 1) ? hbuf0 : hbuf1;

    gru_kernel<<<dim3(HID / 16, BB / 16), 32, 0, stream>>>(
        gates + (size_t)t * BB * GDIM, hc, W_hh, b_hh, hn);

    attn_kernel<<<BB, 256, 0, stream>>>(hn, route_outputs, route_mask, out_id,
                                        wtd, t);

    fc1_kernel<<<dim3(FC1 / 16, BB / 32), 32, 0, stream>>>(hn, wtd, W_fc1,
                                                           b_fc1, hid1);

    fc2_kernel<<<BB / 8, 256, 0, stream>>>(hid1, W_fc2, b_fc2, trg_len,
                                           out_rate, t);
  }
}